// OptimizationModel_49211735278140
// MI455X (gfx1250) — compile-verified
//
#include <hip/hip_runtime.h>
#include <math.h>

typedef float v2f __attribute__((ext_vector_type(2)));
typedef float v8f __attribute__((ext_vector_type(8)));

#define O_OBJ 4
#define NPTS  3072
#define KNN   10
#define TOLC  0.01f
#define AS3 __attribute__((address_space(3)))

// ---------------------------------------------------------------------------
// Kernel 1: per-object transform  T = inv(T_plane) @ T_obj  (rigid inverse)
// tf[o*12 + 0..8] = R row-major, tf[o*12 + 9..11] = t
// ---------------------------------------------------------------------------
__global__ void k_transforms(const float* __restrict__ T_obj,
                             const float* __restrict__ T_plane,
                             float* __restrict__ tf) {
    int o = threadIdx.x;
    if (o >= O_OBJ) return;
    float Rp[3][3], tp[3];
#pragma unroll
    for (int i = 0; i < 3; ++i) {
#pragma unroll
        for (int j = 0; j < 3; ++j) Rp[i][j] = T_plane[4 * i + j];
        tp[i] = T_plane[4 * i + 3];
    }
    const float* To = T_obj + o * 16;
    float* out = tf + o * 12;
    // R = Rp^T * Ro ; t = Rp^T * (to - tp)
#pragma unroll
    for (int i = 0; i < 3; ++i) {
#pragma unroll
        for (int j = 0; j < 3; ++j) {
            float s = 0.f;
#pragma unroll
            for (int kk = 0; kk < 3; ++kk) s += Rp[kk][i] * To[4 * kk + j];
            out[3 * i + j] = s;
        }
        float s = 0.f;
#pragma unroll
        for (int kk = 0; kk < 3; ++kk) s += Rp[kk][i] * (To[4 * kk + 3] - tp[kk]);
        out[9 + i] = s;
    }
}

// ---------------------------------------------------------------------------
// Kernel 2: pip4[o][n] = (x, y, z, x^2+y^2+z^2) in plane frame
// ---------------------------------------------------------------------------
__global__ void k_points(const float* __restrict__ points,
                         const float* __restrict__ tf,
                         float4* __restrict__ pip4) {
    int i = blockIdx.x * blockDim.x + threadIdx.x;
    if (i >= O_OBJ * NPTS) return;
    int o = i / NPTS;
    const float* p = points + (size_t)i * 6;
    const float* t = tf + o * 12;
    float x = t[0] * p[0] + t[1] * p[1] + t[2] * p[2] + t[9];
    float y = t[3] * p[0] + t[4] * p[1] + t[5] * p[2] + t[10];
    float z = t[6] * p[0] + t[7] * p[1] + t[8] * p[2] + t[11];
    float4 v;
    v.x = x; v.y = y; v.z = z; v.w = x * x + y * y + z * z;
    pip4[i] = v;
}

// ---------------------------------------------------------------------------
// Kernel 3: per query point: kNN over each other object via WMMA distance
// tiles fed from an LDS-staged candidate set (async copy, ASYNCcnt),
// inside test, signed min distance vs plane.
// Wave layout: lane owns query n = base + (lane&15); WMMA D tile gives that
// lane candidate rows M = v + 8*(lane>=16)  (C/D layout: N = lane%16).
// ---------------------------------------------------------------------------
__global__ __launch_bounds__(256) void k_main(const float* __restrict__ points,
                                              const float4* __restrict__ pip4,
                                              float* __restrict__ out) {
    __shared__ float4 smem[NPTS];   // 48 KB candidate stage (of 320 KB WGP LDS)

    const int lane = threadIdx.x & 31;
    const int wave = threadIdx.x >> 5;
    const int b    = blockIdx.y;
    const int nq   = blockIdx.x * 128 + wave * 16 + (lane & 15);
    const bool hi  = lane >= 16;

    const float4 q = pip4[b * NPTS + nq];
    const float qr = q.w;

    // B operand (4x16, 2 VGPRs): lanes 0-15 hold K=0/K=1, lanes 16-31 K=2/K=3
    v2f bm;
    bm[0] = hi ? (-2.0f * q.z) : (-2.0f * q.x);
    bm[1] = hi ? 1.0f          : (-2.0f * q.y);

    float dmin = INFINITY;

    for (int o = 0; o < O_OBJ; ++o) {
        if (o == b) continue;                 // uniform across the block
        const float4* cp = pip4 + o * NPTS;

        // --- Stage object o's pip4 into LDS with gfx1250 async copies -----
        __syncthreads();                      // previous o's consumers done
#pragma unroll
        for (int i = 0; i < NPTS / 256; ++i) {
            int idx = (int)threadIdx.x + i * 256;
            unsigned loff =
                (unsigned)(size_t)(AS3 float4*)(smem + idx);
            unsigned long long ga = (unsigned long long)(size_t)(cp + idx);
            asm volatile("global_load_async_to_lds_b128 %0, %1, off"
                         :: "v"(loff), "v"(ga)
                         : "memory");
        }
        asm volatile("s_wait_asynccnt 0x0" ::: "memory");
        __syncthreads();                      // LDS tile visible block-wide

        float bd[KNN]; int bix[KNN];
#pragma unroll
        for (int j = 0; j < KNN; ++j) { bd[j] = INFINITY; bix[j] = 0; }

        for (int mbase = 0; mbase < NPTS; mbase += 16) {
            // A operand (16x4, 2 VGPRs): lane holds candidate m = mbase+lane%16,
            // K=0,1 in low lanes (x,y), K=2,3 in high lanes (z,r). From LDS.
            float4 c = smem[mbase + (lane & 15)];
            v2f am;
            am[0] = hi ? c.z : c.x;
            am[1] = hi ? c.w : c.y;
            v8f acc = {0.f, 0.f, 0.f, 0.f, 0.f, 0.f, 0.f, 0.f};
            acc = __builtin_amdgcn_wmma_f32_16x16x4_f32(
                false, am, false, bm, (short)0, acc, false, false);

            const int moff = mbase + (hi ? 8 : 0);
#pragma unroll
            for (int v = 0; v < 8; ++v) {
                float d = acc[v] + qr;   // r_n - 2 dot + r_m
                int  mi = moff + v;
                if (d < bd[KNN - 1]) {   // divergent fast-reject; no WMMA inside
                    float cd = d; int ci = mi;
#pragma unroll
                    for (int j = 0; j < KNN; ++j) {
                        bool ls = cd < bd[j];
                        float td = bd[j]; int ti = bix[j];
                        bd[j]  = ls ? cd : td;
                        bix[j] = ls ? ci : ti;
                        cd = ls ? td : cd;
                        ci = ls ? ti : ci;
                    }
                }
            }
        }

        // Merge the two half-wave lists (lane n <-> lane n+16). Snapshot first
        // so the shuffled values are the partner's unmodified list.
        float sdl[KNN]; int sil[KNN];
#pragma unroll
        for (int j = 0; j < KNN; ++j) { sdl[j] = bd[j]; sil[j] = bix[j]; }
#pragma unroll
        for (int j = 0; j < KNN; ++j) {
            float od = __shfl_xor(sdl[j], 16, 32);
            int   oi = __shfl_xor(sil[j], 16, 32);
            float cd = od; int ci = oi;
#pragma unroll
            for (int jj = 0; jj < KNN; ++jj) {
                bool ls = cd < bd[jj];
                float td = bd[jj]; int ti = bix[jj];
                bd[jj]  = ls ? cd : td;
                bix[jj] = ls ? ci : ti;
                cd = ls ? td : cd;
                ci = ls ? ti : ci;
            }
        }

        // Inside test: count neighbors whose normal points along (near - q).
        // Neighbor positions come from the LDS stage; normals from global.
        int cnt = 0;
#pragma unroll
        for (int j = 0; j < KNN; ++j) {
            int idx = bix[j];
            float4 np4 = smem[idx];
            const float* nr = points + ((size_t)(o * NPTS + idx)) * 6 + 3;
            float gx = np4.x - q.x, gy = np4.y - q.y, gz = np4.z - q.z;
            float inv = 1.0f / (sqrtf(gx * gx + gy * gy + gz * gz) + 1e-12f);
            float dp = (nr[0] * gx + nr[1] * gy + nr[2] * gz) * inv;
            cnt += (dp > 0.0f) ? 1 : 0;
        }
        float d0 = sqrtf(fmaxf(bd[0], 0.0f));
        if (cnt > 8) d0 = -d0;           // sum(insides) > 0.8*k  (k=10)
        dmin = fminf(dmin, d0);
    }

    float sd = fminf(q.z, dmin);         // plane_sd = z in plane frame
    if (!hi) {
        out[b * NPTS + nq] = sd;
        out[O_OBJ * NPTS + b * NPTS + nq] = (sd < -TOLC) ? 1.0f : 0.0f;
    }
}

// ---------------------------------------------------------------------------
extern "C" void kernel_launch(void* const* d_in, const int* in_sizes, int n_in,
                              void* d_out, int out_size, void* d_ws, size_t ws_size,
                              hipStream_t stream) {
    const float* points  = (const float*)d_in[0];   // (4,3072,6)
    const float* T_obj   = (const float*)d_in[1];   // (4,4,4)
    const float* T_plane = (const float*)d_in[2];   // (4,4)
    // d_in[3] = k (==10, hardcoded)

    float*  tf   = (float*)d_ws;                    // 48 floats (pad to 64)
    float4* pip4 = (float4*)((float*)d_ws + 64);    // 4*3072 float4

    k_transforms<<<1, 32, 0, stream>>>(T_obj, T_plane, tf);
    k_points<<<(O_OBJ * NPTS + 255) / 256, 256, 0, stream>>>(points, tf, pip4);

    dim3 grid(NPTS / 128, O_OBJ);
    k_main<<<grid, 256, 0, stream>>>(points, pip4, (float*)d_out);
}